// GNNQNetwork_90409061581029
// MI455X (gfx1250) — compile-verified
//
#include <hip/hip_runtime.h>
#include <stdint.h>

typedef __attribute__((ext_vector_type(2))) float v2f;
typedef __attribute__((ext_vector_type(8))) float v8f;

#define HDIM 128
#define TM_TILES 8   // M tiles (of 16 rows) processed per block by the h128 GEMM

// ---------------------------------------------------------------------------
// Node encoder: h = x @ W + b   (K features is small: 8 / 13 / 4)
// ---------------------------------------------------------------------------
__global__ __launch_bounds__(256)
void encoder_kernel(const float* __restrict__ x, const float* __restrict__ W,
                    const float* __restrict__ b, float* __restrict__ h,
                    int N, int Kf) {
    int idx = blockIdx.x * blockDim.x + threadIdx.x;
    if (idx >= N * HDIM) return;
    int n = idx >> 7;
    int c = idx & (HDIM - 1);
    const float* xr = x + (size_t)n * Kf;
    float s = b[c];
    for (int k = 0; k < Kf; ++k) s += xr[k] * W[k * HDIM + c];
    h[idx] = s;
}

// wa = W @ a : folds attention vector through the GAT weight so the
// destination-side score needs no full GEMM (ed = h_dst . (W @ a_d)).
__global__ void matvec128(const float* __restrict__ W, const float* __restrict__ a,
                          float* __restrict__ out) {
    int k = threadIdx.x;   // 128 threads
    float s = 0.0f;
    #pragma unroll 8
    for (int c = 0; c < HDIM; ++c) s += W[k * HDIM + c] * a[c];
    out[k] = s;
}

// es[n] = dot(h[n], wa)
__global__ __launch_bounds__(256)
void node_dot(const float* __restrict__ h, const float* __restrict__ v,
              float* __restrict__ out, int N) {
    int n = blockIdx.x * blockDim.x + threadIdx.x;
    if (n >= N) return;
    const float4* hr = (const float4*)(h + (size_t)n * HDIM);
    const float4* vv = (const float4*)v;
    float s = 0.0f;
    #pragma unroll
    for (int i = 0; i < HDIM / 4; ++i) {
        float4 a = hr[i], b = vv[i];
        s += a.x * b.x + a.y * b.y + a.z * b.z + a.w * b.w;
    }
    out[n] = s;
}

// ---------------------------------------------------------------------------
// Specialized fp32 WMMA GEMM for the node transforms: C[M,128] = A[M,128] @ B[128,128]
//
// 8 waves/block, wave w owns a 16-col strip. Each wave keeps ALL of its B
// fragments (32 x v2f = 64 VGPRs) register-resident and sweeps TM_TILES M
// tiles, so the inner loop is exactly 1 global_load_b64 (immediate offset)
// per v_wmma_f32_16x16x4_f32.
//
// ISA layout (7.12.2): A 16x4: lanes 0-15 K={k,k+1}, lanes 16-31 K={k+2,k+3};
// B 4x16 mirrored; C/D: VGPR j -> row j (lanes 0-15) / row j+8 (lanes 16-31).
// Tile-loop bounds are block-uniform and A rows are clamped, so EXEC is all-1s
// at every WMMA; only the stores are predicated.
// ---------------------------------------------------------------------------
__global__ __launch_bounds__(256)
void wmma_gemm_h128_reg(const float* __restrict__ A, const float* __restrict__ B,
                        float* __restrict__ C, int M) {
    const int wave = threadIdx.x >> 5;
    const int lane = threadIdx.x & 31;
    const int half = lane >> 4;        // 0: lanes 0-15, 1: lanes 16-31
    const int lrow = lane & 15;
    const int n0 = wave * 16;          // 8 waves cover all 128 columns

    // one-time B fragment load; stays in VGPRs for the whole block sweep
    v2f breg[32];
    {
        const float* Bp = B + 2 * half * HDIM + n0 + lrow;
        #pragma unroll
        for (int k = 0; k < 32; ++k) {
            breg[k].x = Bp[(4 * k) * HDIM];
            breg[k].y = Bp[(4 * k + 1) * HDIM];
        }
    }

    const int mbase = blockIdx.x * (16 * TM_TILES);
    #pragma unroll 1
    for (int t = 0; t < TM_TILES; ++t) {
        int m0 = mbase + t * 16;
        if (m0 >= M) break;            // block-uniform: EXEC stays full
        int arow = m0 + lrow;
        if (arow >= M) arow = M - 1;   // clamp; stores predicated below
        const float* Aptr = A + (size_t)arow * HDIM + 2 * half;

        v8f acc = {};
        #pragma unroll
        for (int k = 0; k < 32; ++k) {
            v2f a;
            a.x = Aptr[4 * k];
            a.y = Aptr[4 * k + 1];
            acc = __builtin_amdgcn_wmma_f32_16x16x4_f32(
                false, a, false, breg[k], (short)0, acc, false, false);
        }

        #pragma unroll
        for (int j = 0; j < 8; ++j) {
            int r = m0 + j + 8 * half;
            if (r < M) C[(size_t)r * HDIM + n0 + lrow] = acc[j];
        }
    }
}

// ---------------------------------------------------------------------------
// Generic fp32 WMMA GEMM (used only for the small output head, with bias).
// N must be a multiple of 128.
// ---------------------------------------------------------------------------
__global__ __launch_bounds__(256)
void wmma_gemm_f32(const float* __restrict__ A, const float* __restrict__ B,
                   const float* __restrict__ bias, float* __restrict__ C,
                   int M, int K, int N) {
    int wave = threadIdx.x >> 5;
    int lane = threadIdx.x & 31;
    int half = lane >> 4;
    int lrow = lane & 15;
    int m0 = blockIdx.x * 16;
    int n0 = (blockIdx.y * 8 + wave) * 16;

    int arow = m0 + lrow;
    if (arow >= M) arow = M - 1;
    const float* Aptr = A + (size_t)arow * K;

    v8f acc = {};
    for (int k0 = 0; k0 < K; k0 += 4) {
        int ka = k0 + 2 * half;
        v2f a;
        a.x = Aptr[ka];
        a.y = Aptr[ka + 1];
        v2f b;
        b.x = B[(size_t)ka * N + n0 + lrow];
        b.y = B[(size_t)(ka + 1) * N + n0 + lrow];
        acc = __builtin_amdgcn_wmma_f32_16x16x4_f32(
            false, a, false, b, (short)0, acc, false, false);
    }

    float bv = bias ? bias[n0 + lrow] : 0.0f;
    #pragma unroll
    for (int j = 0; j < 8; ++j) {
        int r = m0 + j + 8 * half;
        if (r < M) C[(size_t)r * N + n0 + lrow] = acc[j] + bv;
    }
}

// ---------------------------------------------------------------------------
// Segment softmax over edges (monotonic uint encoding for float atomicMax)
// ---------------------------------------------------------------------------
__device__ __forceinline__ unsigned fenc(float f) {
    unsigned u = __float_as_uint(f);
    return (u & 0x80000000u) ? ~u : (u | 0x80000000u);
}
__device__ __forceinline__ float fdec(unsigned u) {
    return (u & 0x80000000u) ? __uint_as_float(u & 0x7fffffffu)
                             : __uint_as_float(~u);
}

__global__ __launch_bounds__(256)
void edge_score_max(const float* __restrict__ es, const float* __restrict__ ed,
                    const int* __restrict__ src, const int* __restrict__ dst,
                    float* __restrict__ ebuf, unsigned* __restrict__ menc, int E) {
    int k = blockIdx.x * blockDim.x + threadIdx.x;
    if (k >= E) return;
    float e = es[src[k]] + ed[dst[k]];
    e = (e > 0.0f) ? e : 0.2f * e;          // leaky_relu(0.2)
    ebuf[k] = e;
    atomicMax(&menc[dst[k]], fenc(e));
}

__global__ __launch_bounds__(256)
void edge_denom(const float* __restrict__ ebuf, const unsigned* __restrict__ menc,
                float* __restrict__ denom, const int* __restrict__ dst, int E) {
    int k = blockIdx.x * blockDim.x + threadIdx.x;
    if (k >= E) return;
    int d = dst[k];
    atomicAdd(&denom[d], __expf(ebuf[k] - fdec(menc[d])));
}

// One wave per edge; each lane handles 4 channels. Accumulates alpha*hs[src]
// directly into the per-graph pooled sum (mean-pool is linear, so per-node
// outputs never materialize; biases fold in analytically later).
__global__ __launch_bounds__(256)
void edge_scatter(const float* __restrict__ hs, const float* __restrict__ ebuf,
                  const unsigned* __restrict__ menc, const float* __restrict__ denom,
                  const int* __restrict__ src, const int* __restrict__ dst,
                  const int* __restrict__ batch_dst, float* __restrict__ pool, int E) {
    int gid = blockIdx.x * blockDim.x + threadIdx.x;
    int k = gid >> 5;
    if (k >= E) return;
    int lane = gid & 31;
    int d = dst[k];
    float alpha = __expf(ebuf[k] - fdec(menc[d])) / fmaxf(denom[d], 1e-16f);
    int b = batch_dst[d];
    const float4* hv = (const float4*)(hs + (size_t)src[k] * HDIM);
    float4 v = hv[lane];
    float* p = pool + (size_t)b * HDIM + lane * 4;
    atomicAdd(p + 0, alpha * v.x);
    atomicAdd(p + 1, alpha * v.y);
    atomicAdd(p + 2, alpha * v.z);
    atomicAdd(p + 3, alpha * v.w);
}

__global__ __launch_bounds__(256)
void count_nodes(const int* __restrict__ batch, float* __restrict__ cnt, int N) {
    int n = blockIdx.x * blockDim.x + threadIdx.x;
    if (n < N) atomicAdd(&cnt[batch[n]], 1.0f);
}

// ---------------------------------------------------------------------------
// Assemble x = [mean_t | mean_rv | mean_r | player_mlp]  -> [B, 512]
// mean = pooled_sum / max(count,1) + (bias_et1 + bias_et2)
// ---------------------------------------------------------------------------
__global__ __launch_bounds__(256)
void build_x(const float* __restrict__ pool_t, const float* __restrict__ pool_rv,
             const float* __restrict__ pool_r,
             const float* __restrict__ cnt_t, const float* __restrict__ cnt_rv,
             const float* __restrict__ cnt_r,
             const float* __restrict__ bt1, const float* __restrict__ bt2,
             const float* __restrict__ brv1, const float* __restrict__ brv2,
             const float* __restrict__ br1, const float* __restrict__ br2,
             const float* __restrict__ xp, const float* __restrict__ fc1W,
             const float* __restrict__ fc1b, const float* __restrict__ fc2W,
             const float* __restrict__ fc2b,
             float* __restrict__ x, int B) {
    int idx = blockIdx.x * blockDim.x + threadIdx.x;
    if (idx >= B * 512) return;
    int b = idx >> 9;
    int c = idx & 511;
    float v;
    if (c < 128) {
        v = pool_t[b * HDIM + c] / fmaxf(cnt_t[b], 1.0f) + bt1[c] + bt2[c];
    } else if (c < 256) {
        int cc = c - 128;
        v = pool_rv[b * HDIM + cc] / fmaxf(cnt_rv[b], 1.0f) + brv1[cc] + brv2[cc];
    } else if (c < 384) {
        int cc = c - 256;
        v = pool_r[b * HDIM + cc] / fmaxf(cnt_r[b], 1.0f) + br1[cc] + br2[cc];
    } else {
        int cc = c - 384;                       // player MLP output channel
        const float* xr = xp + (size_t)b * 64;
        float s2 = fc2b[cc];
        for (int j = 0; j < 128; ++j) {
            float s1 = fc1b[j];
            #pragma unroll 8
            for (int i = 0; i < 64; ++i) s1 += xr[i] * fc1W[i * 128 + j];
            s1 = fmaxf(s1, 0.0f);
            s2 += s1 * fc2W[j * 128 + cc];
        }
        v = fmaxf(s2, 0.0f);
    }
    x[idx] = v;
}

// ---------------------------------------------------------------------------
// Host-side orchestration
// ---------------------------------------------------------------------------
static inline int cdiv(long long a, long long b) { return (int)((a + b - 1) / b); }

extern "C" void kernel_launch(void* const* d_in, const int* in_sizes, int n_in,
                              void* d_out, int out_size, void* d_ws, size_t ws_size,
                              hipStream_t stream) {
    (void)n_in; (void)ws_size;
    const float* x_tile   = (const float*)d_in[0];
    const float* x_rv     = (const float*)d_in[1];
    const float* x_road   = (const float*)d_in[2];
    const float* x_player = (const float*)d_in[3];

    const int NT  = in_sizes[0] / 8;
    const int NRV = in_sizes[1] / 13;
    const int NR  = in_sizes[2] / 4;
    const int B   = in_sizes[3] / 64;
    const int E   = in_sizes[40];

    const float* enc_t_W  = (const float*)d_in[4];
    const float* enc_t_b  = (const float*)d_in[5];
    const float* enc_rv_W = (const float*)d_in[6];
    const float* enc_rv_b = (const float*)d_in[7];
    const float* enc_r_W  = (const float*)d_in[8];
    const float* enc_r_b  = (const float*)d_in[9];

    const float* gW[6];  const float* gAs[6]; const float* gAd[6]; const float* gB[6];
    for (int g = 0; g < 6; ++g) {
        gW[g]  = (const float*)d_in[10 + 4 * g + 0];
        gAs[g] = (const float*)d_in[10 + 4 * g + 1];
        gAd[g] = (const float*)d_in[10 + 4 * g + 2];
        gB[g]  = (const float*)d_in[10 + 4 * g + 3];
    }
    const float* fc1W = (const float*)d_in[34];
    const float* fc1b = (const float*)d_in[35];
    const float* fc2W = (const float*)d_in[36];
    const float* fc2b = (const float*)d_in[37];
    const float* outW = (const float*)d_in[38];
    const float* outb = (const float*)d_in[39];

    const int* eSrc[6]; const int* eDst[6];
    for (int g = 0; g < 6; ++g) {
        eSrc[g] = (const int*)d_in[40 + 2 * g + 0];
        eDst[g] = (const int*)d_in[40 + 2 * g + 1];
    }
    const int* batch_tile = (const int*)d_in[52];
    const int* batch_rv   = (const int*)d_in[53];
    const int* batch_road = (const int*)d_in[54];

    // ---- workspace carve (256B aligned) -------------------------------
    char* wsp = (char*)d_ws;
    size_t off = 0;
    auto carve = [&](size_t bytes) -> char* {
        char* p = wsp + off;
        off += (bytes + 255) & ~(size_t)255;
        return p;
    };
    float*    h_t    = (float*)carve((size_t)NT  * HDIM * 4);
    float*    h_rv   = (float*)carve((size_t)NRV * HDIM * 4);
    float*    h_r    = (float*)carve((size_t)NR  * HDIM * 4);
    float*    hs     = (float*)carve((size_t)NR  * HDIM * 4);   // max Ns
    float*    es     = (float*)carve((size_t)NR * 4);
    float*    ed     = (float*)carve((size_t)NR * 4);
    unsigned* menc   = (unsigned*)carve((size_t)NR * 4);
    float*    denom  = (float*)carve((size_t)NR * 4);
    float*    ebuf   = (float*)carve((size_t)E * 4);
    float*    wa_s   = (float*)carve(HDIM * 4);
    float*    wa_d   = (float*)carve(HDIM * 4);
    float*    pool_t = (float*)carve((size_t)B * HDIM * 4);
    float*    pool_rv= (float*)carve((size_t)B * HDIM * 4);
    float*    pool_r = (float*)carve((size_t)B * HDIM * 4);
    float*    cnt_t  = (float*)carve((size_t)B * 4);
    float*    cnt_rv = (float*)carve((size_t)B * 4);
    float*    cnt_r  = (float*)carve((size_t)B * 4);
    float*    x_cat  = (float*)carve((size_t)B * 512 * 4);

    // ---- node encoders ------------------------------------------------
    encoder_kernel<<<cdiv((long long)NT  * HDIM, 256), 256, 0, stream>>>(x_tile, enc_t_W, enc_t_b, h_t, NT, 8);
    encoder_kernel<<<cdiv((long long)NRV * HDIM, 256), 256, 0, stream>>>(x_rv, enc_rv_W, enc_rv_b, h_rv, NRV, 13);
    encoder_kernel<<<cdiv((long long)NR  * HDIM, 256), 256, 0, stream>>>(x_road, enc_r_W, enc_r_b, h_r, NR, 4);

    // ---- pooled accumulators + counts --------------------------------
    hipMemsetAsync(pool_t,  0, (size_t)B * HDIM * 4, stream);
    hipMemsetAsync(pool_rv, 0, (size_t)B * HDIM * 4, stream);
    hipMemsetAsync(pool_r,  0, (size_t)B * HDIM * 4, stream);
    hipMemsetAsync(cnt_t,   0, (size_t)B * 4, stream);
    hipMemsetAsync(cnt_rv,  0, (size_t)B * 4, stream);
    hipMemsetAsync(cnt_r,   0, (size_t)B * 4, stream);
    count_nodes<<<cdiv(NT, 256), 256, 0, stream>>>(batch_tile, cnt_t, NT);
    count_nodes<<<cdiv(NRV, 256), 256, 0, stream>>>(batch_rv, cnt_rv, NRV);
    count_nodes<<<cdiv(NR, 256), 256, 0, stream>>>(batch_road, cnt_r, NR);

    // ---- GAT passes (order: rv2t, t2rv, r2rv, rv2r, t2r, r2t) ---------
    const float* hsrc[6]   = { h_rv, h_t, h_r, h_rv, h_t, h_r };
    const int    nsrc[6]   = { NRV,  NT,  NR,  NRV,  NT,  NR  };
    const float* hdst[6]   = { h_t,  h_rv, h_rv, h_r, h_r, h_t };
    const int    ndst[6]   = { NT,   NRV, NRV, NR,  NR,  NT  };
    const int*   bdst[6]   = { batch_tile, batch_rv, batch_rv, batch_road, batch_road, batch_tile };
    float*       pdst[6]   = { pool_t, pool_rv, pool_rv, pool_r, pool_r, pool_t };

    for (int g = 0; g < 6; ++g) {
        const int Ns = nsrc[g], Nd = ndst[g];
        // fold attention vectors through W
        matvec128<<<1, 128, 0, stream>>>(gW[g], gAs[g], wa_s);
        matvec128<<<1, 128, 0, stream>>>(gW[g], gAd[g], wa_d);
        // per-node attention scores (no dst-side GEMM needed)
        node_dot<<<cdiv(Ns, 256), 256, 0, stream>>>(hsrc[g], wa_s, es, Ns);
        node_dot<<<cdiv(Nd, 256), 256, 0, stream>>>(hdst[g], wa_d, ed, Nd);
        // hs = h_src @ W  (fp32 WMMA, register-resident B)
        wmma_gemm_h128_reg<<<cdiv(Ns, 16 * TM_TILES), 256, 0, stream>>>(
            hsrc[g], gW[g], hs, Ns);
        // segment softmax
        hipMemsetAsync(menc,  0, (size_t)Nd * 4, stream);
        hipMemsetAsync(denom, 0, (size_t)Nd * 4, stream);
        edge_score_max<<<cdiv(E, 256), 256, 0, stream>>>(es, ed, eSrc[g], eDst[g], ebuf, menc, E);
        edge_denom<<<cdiv(E, 256), 256, 0, stream>>>(ebuf, menc, denom, eDst[g], E);
        // weighted message scatter, fused with batch mean-pool accumulation
        edge_scatter<<<cdiv((long long)E * 32, 256), 256, 0, stream>>>(
            hs, ebuf, menc, denom, eSrc[g], eDst[g], bdst[g], pdst[g], E);
    }

    // ---- assemble [B,512] feature vector (means + biases + player MLP)
    build_x<<<cdiv((long long)B * 512, 256), 256, 0, stream>>>(
        pool_t, pool_rv, pool_r, cnt_t, cnt_rv, cnt_r,
        gB[0], gB[5],      // tile: rv2t_b + r2t_b
        gB[1], gB[2],      // rv:   t2rv_b + r2rv_b
        gB[3], gB[4],      // road: rv2r_b + t2r_b
        x_player, fc1W, fc1b, fc2W, fc2b,
        x_cat, B);

    // ---- output head: [B,512] @ [512,512] + out_b  (fp32 WMMA) --------
    (void)out_size;
    wmma_gemm_f32<<<dim3(cdiv(B, 16), 4), 256, 0, stream>>>(
        x_cat, outW, outb, (float*)d_out, B, 512, 512);
}